// DecentralizedAttentionLayer_28106265985634
// MI455X (gfx1250) — compile-verified
//
#include <hip/hip_runtime.h>
#include <hip/hip_bf16.h>
#include <math.h>

#define NNODES 100000
#define NEDGES 1600000
#define DIM 128
#define EPS 1e-6f
#define NEG_SLOPE 0.2f

typedef __attribute__((ext_vector_type(2))) float v2f;
typedef __attribute__((ext_vector_type(8))) float v8f;

__device__ __forceinline__ float wave_sum(float v) {
#pragma unroll
    for (int m = 16; m > 0; m >>= 1) v += __shfl_xor(v, m, 32);
    return v;
}
__device__ __forceinline__ float wave_max(float v) {
#pragma unroll
    for (int m = 16; m > 0; m >>= 1) v = fmaxf(v, __shfl_xor(v, m, 32));
    return v;
}

// ---------------------------------------------------------------------------
// Kernel 0: fold w1/w2 GEMMs into vectors.  sum(q@W+b, axis=1) == q.(W@1) + sum(b)
// ---------------------------------------------------------------------------
__global__ void prep_kernel(const float* __restrict__ w1_W, const float* __restrict__ w1_b,
                            const float* __restrict__ w2_W, const float* __restrict__ w2_b,
                            float* __restrict__ u1, float* __restrict__ u2,
                            float* __restrict__ c12) {
    int d = threadIdx.x;
    if (d < DIM) {
        float s1 = 0.f, s2 = 0.f;
        for (int j = 0; j < DIM; ++j) {
            s1 += w1_W[d * DIM + j];
            s2 += w2_W[d * DIM + j];
        }
        u1[d] = s1;
        u2[d] = s2;
    }
    if (d == 0) {
        float b1 = 0.f, b2 = 0.f;
        for (int j = 0; j < DIM; ++j) { b1 += w1_b[j]; b2 += w2_b[j]; }
        c12[0] = b1;
        c12[1] = b2;
    }
}

// ---------------------------------------------------------------------------
// Kernel 1: per-node layernorm + scalar scores (one wave32 per node, 4 dims/lane)
// ---------------------------------------------------------------------------
__global__ void __launch_bounds__(256)
node_norm_kernel(const float* __restrict__ query,
                 const float* __restrict__ g, const float* __restrict__ b,
                 const float* __restrict__ u1, const float* __restrict__ u2,
                 const float* __restrict__ c12,
                 float* __restrict__ qn,
                 float* __restrict__ sum1, float* __restrict__ sum2) {
    int lane = threadIdx.x & 31;
    int node = blockIdx.x * 8 + (threadIdx.x >> 5);
    if (node >= NNODES) return;

    const float4 x = *(const float4*)(query + (size_t)node * DIM + lane * 4);
    float mean = wave_sum(x.x + x.y + x.z + x.w) * (1.0f / DIM);
    float cx = x.x - mean, cy = x.y - mean, cz = x.z - mean, cw = x.w - mean;
    float var = wave_sum(cx * cx + cy * cy + cz * cz + cw * cw) * (1.0f / DIM);
    float rstd = rsqrtf(var + EPS);

    const float4 gg = *(const float4*)(g + lane * 4);
    const float4 bb = *(const float4*)(b + lane * 4);
    float4 q4;
    q4.x = gg.x * cx * rstd + bb.x;
    q4.y = gg.y * cy * rstd + bb.y;
    q4.z = gg.z * cz * rstd + bb.z;
    q4.w = gg.w * cw * rstd + bb.w;
    *(float4*)(qn + (size_t)node * DIM + lane * 4) = q4;

    const float4 a1 = *(const float4*)(u1 + lane * 4);
    const float4 a2 = *(const float4*)(u2 + lane * 4);
    float d1 = wave_sum(q4.x * a1.x + q4.y * a1.y + q4.z * a1.z + q4.w * a1.w);
    float d2 = wave_sum(q4.x * a2.x + q4.y * a2.y + q4.z * a2.z + q4.w * a2.w);
    if (lane == 0) {
        sum1[node] = tanhf(d1 + c12[0]);
        sum2[node] = tanhf(d2 + c12[1]);
    }
}

// ---------------------------------------------------------------------------
// Kernel 2: row_ptr via binary search (adj_row is sorted)
// ---------------------------------------------------------------------------
__global__ void row_ptr_kernel(const int* __restrict__ adj_row, int* __restrict__ row_ptr) {
    int i = blockIdx.x * blockDim.x + threadIdx.x;
    if (i > NNODES) return;
    int lo = 0, hi = NEDGES;
    while (lo < hi) {
        int mid = (lo + hi) >> 1;
        if (adj_row[mid] < i) lo = mid + 1; else hi = mid;
    }
    row_ptr[i] = lo;
}

// ---------------------------------------------------------------------------
// Kernel 3: value = qn @ w_W + w_b  via V_WMMA_F32_16X16X4_F32
//
// w_W staged in LDS in K-PAIRED layout:  wlds[(k/2)*128 + c] = {W[k][c], W[k+1][c]}
// so each lane's B fragment (needs dwords {W[kk][c], W[kk+1][c]}, kk even) is a
// single aligned ds_load_b64 directly into the even VGPR pair -> no repack movs.
//
// A 16x4 f32 layout: lane l -> M = l&15; VGPR0/1 hold K = k0+2*(l>>4) .. +1
// C/D 16x16 f32 layout: VGPR r -> M = r + 8*(l>>4), N = l&15
// Each wave owns a 16x128 output strip (8 v8f accumulators).
// ---------------------------------------------------------------------------
__global__ void __launch_bounds__(256)
value_gemm_wmma(const float* __restrict__ qn, const float* __restrict__ wW,
                const float* __restrict__ wb, float* __restrict__ val) {
    __shared__ float2 wlds[(DIM / 2) * DIM];  // 65536 bytes
    for (int i = threadIdx.x; i < (DIM / 2) * DIM; i += 256) {
        const int k2 = i >> 7;        // pair index: rows 2*k2, 2*k2+1
        const int c  = i & 127;       // column
        float2 p;
        p.x = wW[(2 * k2) * DIM + c];
        p.y = wW[(2 * k2 + 1) * DIM + c];
        wlds[i] = p;
    }
    __syncthreads();

    const int lane = threadIdx.x & 31;
    const int n = lane & 15;
    const int hl = lane >> 4;
    const int rowBase = (blockIdx.x * 8 + (threadIdx.x >> 5)) * 16;
    if (rowBase >= NNODES) return;

    v8f acc[8];
#pragma unroll
    for (int t = 0; t < 8; ++t) {
        float bias = wb[t * 16 + n];
        acc[t] = (v8f){bias, bias, bias, bias, bias, bias, bias, bias};
    }

    const float* arow = qn + (size_t)(rowBase + n) * DIM;  // A row M = lane&15

    for (int k0 = 0; k0 < DIM; k0 += 4) {
        const int kk = k0 + 2 * hl;
        v2f a;
        a.x = arow[kk];
        a.y = arow[kk + 1];
        const float2* bp = wlds + ((k0 >> 1) + hl) * DIM + n;
#pragma unroll
        for (int t = 0; t < 8; ++t) {
            const float2 bv = bp[t * 16];
            v2f bf;
            bf.x = bv.x;
            bf.y = bv.y;
            acc[t] = __builtin_amdgcn_wmma_f32_16x16x4_f32(
                false, a, false, bf, (short)0, acc[t], false, false);
        }
    }

#pragma unroll
    for (int t = 0; t < 8; ++t) {
#pragma unroll
        for (int r = 0; r < 8; ++r) {
            val[(size_t)(rowBase + r + 8 * hl) * DIM + t * 16 + n] = acc[t][r];
        }
    }
}

// ---------------------------------------------------------------------------
// Kernel 4: per-node edge softmax + SpMM + final layernorm (one wave per node)
// ---------------------------------------------------------------------------
__global__ void __launch_bounds__(256)
attn_kernel(const float* __restrict__ adj_val, const int* __restrict__ adj_col,
            const int* __restrict__ row_ptr,
            const float* __restrict__ sum1, const float* __restrict__ sum2,
            const float* __restrict__ val,
            const float* __restrict__ g2, const float* __restrict__ b2,
            float* __restrict__ out) {
    int lane = threadIdx.x & 31;
    int node = blockIdx.x * 8 + (threadIdx.x >> 5);
    if (node >= NNODES) return;

    const int beg = row_ptr[node];
    const int end = row_ptr[node + 1];
    const float s1 = sum1[node];

    // pass 1: segment max of leaky_relu logits (lanes stride over edges)
    float m = -3.402823466e38f;
    for (int e = beg + lane; e < end; e += 32) {
        float x = adj_val[e] * (s1 + sum2[adj_col[e]]);
        x = x > 0.f ? x : NEG_SLOPE * x;
        m = fmaxf(m, x);
    }
    m = wave_max(m);

    // pass 2: exp-weights + gather-accumulate value rows (all lanes cooperate per edge)
    float ax = 0.f, ay = 0.f, az = 0.f, aw = 0.f;
    float ssum = 0.f;
    for (int e = beg; e < end; ++e) {
        const int c = adj_col[e];
        float x = adj_val[e] * (s1 + sum2[c]);
        x = x > 0.f ? x : NEG_SLOPE * x;
        float w = expf(x - m);
        ssum += w;
        const float4 v = *(const float4*)(val + (size_t)c * DIM + lane * 4);
        ax += w * v.x; ay += w * v.y; az += w * v.z; aw += w * v.w;
    }
    const float inv = ssum > 0.f ? 1.f / ssum : 0.f;
    ax *= inv; ay *= inv; az *= inv; aw *= inv;

    // final layernorm
    float mean = wave_sum(ax + ay + az + aw) * (1.0f / DIM);
    float cx = ax - mean, cy = ay - mean, cz = az - mean, cw = aw - mean;
    float var = wave_sum(cx * cx + cy * cy + cz * cz + cw * cw) * (1.0f / DIM);
    float rstd = rsqrtf(var + EPS);

    const float4 gg = *(const float4*)(g2 + lane * 4);
    const float4 bb = *(const float4*)(b2 + lane * 4);
    float4 o;
    o.x = gg.x * cx * rstd + bb.x;
    o.y = gg.y * cy * rstd + bb.y;
    o.z = gg.z * cz * rstd + bb.z;
    o.w = gg.w * cw * rstd + bb.w;
    *(float4*)(out + (size_t)node * DIM + lane * 4) = o;
}

// ---------------------------------------------------------------------------
extern "C" void kernel_launch(void* const* d_in, const int* in_sizes, int n_in,
                              void* d_out, int out_size, void* d_ws, size_t ws_size,
                              hipStream_t stream) {
    const float* query  = (const float*)d_in[0];
    const float* adjval = (const float*)d_in[1];
    const float* w_W    = (const float*)d_in[2];
    const float* w_b    = (const float*)d_in[3];
    const float* w1_W   = (const float*)d_in[4];
    const float* w1_b   = (const float*)d_in[5];
    const float* w2_W   = (const float*)d_in[6];
    const float* w2_b   = (const float*)d_in[7];
    const float* ln1_g  = (const float*)d_in[8];
    const float* ln1_b  = (const float*)d_in[9];
    const float* ln2_g  = (const float*)d_in[10];
    const float* ln2_b  = (const float*)d_in[11];
    const int*   adjrow = (const int*)d_in[12];
    const int*   adjcol = (const int*)d_in[13];
    float* out = (float*)d_out;

    // workspace layout (floats)
    float* ws   = (float*)d_ws;
    float* qn   = ws;                              // N*DIM
    float* val  = qn + (size_t)NNODES * DIM;       // N*DIM
    float* sum1 = val + (size_t)NNODES * DIM;      // N
    float* sum2 = sum1 + NNODES;                   // N
    float* u1   = sum2 + NNODES;                   // DIM
    float* u2   = u1 + DIM;                        // DIM
    float* c12  = u2 + DIM;                        // 2
    int* row_ptr = (int*)(c12 + 2);                // N+1

    prep_kernel<<<1, 128, 0, stream>>>(w1_W, w1_b, w2_W, w2_b, u1, u2, c12);

    const int nodeBlocks = (NNODES + 7) / 8;
    node_norm_kernel<<<nodeBlocks, 256, 0, stream>>>(
        query, ln1_g, ln1_b, u1, u2, c12, qn, sum1, sum2);

    row_ptr_kernel<<<(NNODES + 1 + 255) / 256, 256, 0, stream>>>(adjrow, row_ptr);

    const int tiles = NNODES / 16;                 // 6250 (exact)
    value_gemm_wmma<<<(tiles + 7) / 8, 256, 0, stream>>>(qn, w_W, w_b, val);

    attn_kernel<<<nodeBlocks, 256, 0, stream>>>(
        adjval, adjcol, row_ptr, sum1, sum2, val, ln2_g, ln2_b, out);
}